// GraphSAGEModel_83623013253774
// MI455X (gfx1250) — compile-verified
//
#include <hip/hip_runtime.h>

typedef float v2f __attribute__((ext_vector_type(2)));
typedef float v8f __attribute__((ext_vector_type(8)));

// ---------------------------------------------------------------- utilities

__global__ void k_count(const long long* __restrict__ ei, float* __restrict__ cnt, int E) {
    int i = blockIdx.x * blockDim.x + threadIdx.x;
    if (i < E) atomicAdd(&cnt[ei[(long long)E + i]], 1.0f);
}

__global__ void k_inv(const float* __restrict__ cnt, float* __restrict__ inv, int n) {
    int i = blockIdx.x * blockDim.x + threadIdx.x;
    if (i < n) inv[i] = 1.0f / fmaxf(cnt[i], 1.0f);
}

// scatter-add F features per edge: agg[dst] += feat[src]
template<int F, int LOGF>
__global__ void k_scatter(const float* __restrict__ feat, const long long* __restrict__ ei,
                          float* __restrict__ agg, int E) {
    long long total = (long long)E << LOGF;
    long long idx = (long long)blockIdx.x * blockDim.x + threadIdx.x;
    if (idx >= total) return;
    int e = (int)(idx >> LOGF);
    int c = (int)(idx & (F - 1));
    long long s = ei[e];
    long long d = ei[(long long)E + e];
    atomicAdd(&agg[d * F + c], feat[s * F + c]);
}

__global__ void k_scatter1(const float* __restrict__ sv, const long long* __restrict__ ei,
                           float* __restrict__ agg, int E) {
    int i = blockIdx.x * blockDim.x + threadIdx.x;
    if (i < E) atomicAdd(&agg[ei[(long long)E + i]], sv[ei[i]]);
}

// ------------------------------------------- fused SAGE layer GEMM via WMMA
// H[r, 0:64] = relu( (Aagg[r]*inv[r]) @ Bl  +  Aself[r] @ Br  + bias )
// One wave computes a 16x64 output tile with V_WMMA_F32_16X16X4_F32.
template<int K1, int K2>
__global__ __launch_bounds__(256)
void sage_gemm_relu(const float* __restrict__ Aagg, const float* __restrict__ inv,
                    const float* __restrict__ Aself,
                    const float* __restrict__ Bl, const float* __restrict__ Br,
                    const float* __restrict__ bias, float* __restrict__ Hout, int n) {
    int wave = (int)(((unsigned)blockIdx.x * blockDim.x + threadIdx.x) >> 5);
    int lane = threadIdx.x & 31;
    int m0 = wave * 16;
    if (m0 >= n) return;                       // wave-uniform: EXEC stays all-ones
    int mrow = m0 + (lane & 15);
    int r = (mrow < n) ? mrow : (n - 1);       // clamp (loads only; stores masked)
    int kh = (lane >> 4) << 1;                 // lanes 0-15 -> K {0,1}; 16-31 -> K {2,3}
    int col = lane & 15;
    float invr = inv[r];

    v8f c0 = {}, c1 = {}, c2 = {}, c3 = {};

    // aggregated-neighbor part: A = Aagg * invr, B = Bl [K1 x 64]
    for (int k = 0; k < K1; k += 4) {
        v2f a = *(const v2f*)(Aagg + (size_t)r * K1 + k + kh);
        a.x *= invr; a.y *= invr;
        v2f b;
        b.x = Bl[(k + kh) * 64 +  0 + col]; b.y = Bl[(k + kh + 1) * 64 +  0 + col];
        c0 = __builtin_amdgcn_wmma_f32_16x16x4_f32(false, a, false, b, (short)0, c0, false, false);
        b.x = Bl[(k + kh) * 64 + 16 + col]; b.y = Bl[(k + kh + 1) * 64 + 16 + col];
        c1 = __builtin_amdgcn_wmma_f32_16x16x4_f32(false, a, false, b, (short)0, c1, false, false);
        b.x = Bl[(k + kh) * 64 + 32 + col]; b.y = Bl[(k + kh + 1) * 64 + 32 + col];
        c2 = __builtin_amdgcn_wmma_f32_16x16x4_f32(false, a, false, b, (short)0, c2, false, false);
        b.x = Bl[(k + kh) * 64 + 48 + col]; b.y = Bl[(k + kh + 1) * 64 + 48 + col];
        c3 = __builtin_amdgcn_wmma_f32_16x16x4_f32(false, a, false, b, (short)0, c3, false, false);
    }
    // self part: A = Aself, B = Br [K2 x 64]
    for (int k = 0; k < K2; k += 4) {
        v2f a = *(const v2f*)(Aself + (size_t)r * K2 + k + kh);
        v2f b;
        b.x = Br[(k + kh) * 64 +  0 + col]; b.y = Br[(k + kh + 1) * 64 +  0 + col];
        c0 = __builtin_amdgcn_wmma_f32_16x16x4_f32(false, a, false, b, (short)0, c0, false, false);
        b.x = Br[(k + kh) * 64 + 16 + col]; b.y = Br[(k + kh + 1) * 64 + 16 + col];
        c1 = __builtin_amdgcn_wmma_f32_16x16x4_f32(false, a, false, b, (short)0, c1, false, false);
        b.x = Br[(k + kh) * 64 + 32 + col]; b.y = Br[(k + kh + 1) * 64 + 32 + col];
        c2 = __builtin_amdgcn_wmma_f32_16x16x4_f32(false, a, false, b, (short)0, c2, false, false);
        b.x = Br[(k + kh) * 64 + 48 + col]; b.y = Br[(k + kh + 1) * 64 + 48 + col];
        c3 = __builtin_amdgcn_wmma_f32_16x16x4_f32(false, a, false, b, (short)0, c3, false, false);
    }

    // store: C/D layout -> VGPR v: lanes 0-15 row m0+v, lanes 16-31 row m0+8+v
    int rbase = m0 + ((lane >> 4) << 3);
    float bi0 = bias[col], bi1 = bias[16 + col], bi2 = bias[32 + col], bi3 = bias[48 + col];
#pragma unroll
    for (int v = 0; v < 8; v++) {
        int row = rbase + v;
        if (row < n) {
            float* o = Hout + (size_t)row * 64 + col;
            o[0]  = fmaxf(c0[v] + bi0, 0.0f);
            o[16] = fmaxf(c1[v] + bi1, 0.0f);
            o[32] = fmaxf(c2[v] + bi2, 0.0f);
            o[48] = fmaxf(c3[v] + bi3, 0.0f);
        }
    }
}

// ----------------------------------- layer 2: per-node dot with W_l2 / W_r2
__global__ void k_lin2(const float* __restrict__ h1, const float* __restrict__ wl,
                       const float* __restrict__ wr, float* __restrict__ sl,
                       float* __restrict__ sr, int n) {
    int w = (int)(((unsigned)blockIdx.x * blockDim.x + threadIdx.x) >> 5);
    int lane = threadIdx.x & 31;
    if (w >= n) return;
    float h0 = h1[(size_t)w * 64 + lane];
    float h32 = h1[(size_t)w * 64 + 32 + lane];
    float al = h0 * wl[lane] + h32 * wl[32 + lane];
    float ar = h0 * wr[lane] + h32 * wr[32 + lane];
#pragma unroll
    for (int off = 16; off > 0; off >>= 1) {
        al += __shfl_down(al, off, 32);
        ar += __shfl_down(ar, off, 32);
    }
    if (lane == 0) { sl[w] = al; sr[w] = ar; }
}

__global__ void k_final(const float* __restrict__ sagg, const float* __restrict__ inv,
                        const float* __restrict__ sr, const float* __restrict__ b2,
                        float* __restrict__ out, int n) {
    int i = blockIdx.x * blockDim.x + threadIdx.x;
    if (i < n) out[i] = sagg[i] * inv[i] + sr[i] + b2[0];
}

// ---------------------------------------------------------------- launcher

extern "C" void kernel_launch(void* const* d_in, const int* in_sizes, int n_in,
                              void* d_out, int out_size, void* d_ws, size_t ws_size,
                              hipStream_t stream) {
    const float*     x   = (const float*)d_in[0];
    const long long* ei  = (const long long*)d_in[1];
    const float*     Wl0 = (const float*)d_in[2];
    const float*     Wr0 = (const float*)d_in[3];
    const float*     b0  = (const float*)d_in[4];
    const float*     Wl1 = (const float*)d_in[5];
    const float*     Wr1 = (const float*)d_in[6];
    const float*     b1  = (const float*)d_in[7];
    const float*     Wl2 = (const float*)d_in[8];
    const float*     Wr2 = (const float*)d_in[9];
    const float*     b2  = (const float*)d_in[10];

    int n = in_sizes[0] / 32;
    int E = in_sizes[1] / 2;

    float* ws    = (float*)d_ws;
    float* cnt   = ws;  ws += n;
    float* inv   = ws;  ws += n;
    float* agg0  = ws;  ws += (size_t)n * 32;
    float* h0    = ws;  ws += (size_t)n * 64;
    float* agg1  = ws;  ws += (size_t)n * 64;
    float* h1    = ws;  ws += (size_t)n * 64;
    float* s2l   = ws;  ws += n;
    float* s2r   = ws;  ws += n;
    float* s2agg = ws;

    hipMemsetAsync(cnt,   0, (size_t)n * sizeof(float), stream);
    hipMemsetAsync(agg0,  0, (size_t)n * 32 * sizeof(float), stream);
    hipMemsetAsync(agg1,  0, (size_t)n * 64 * sizeof(float), stream);
    hipMemsetAsync(s2agg, 0, (size_t)n * sizeof(float), stream);

    k_count<<<(E + 255) / 256, 256, 0, stream>>>(ei, cnt, E);
    k_inv  <<<(n + 255) / 256, 256, 0, stream>>>(cnt, inv, n);

    int gemm_blocks = (int)((((long long)(n + 15) / 16) * 32 + 255) / 256);

    // layer 0: aggregate x (32-d payload) first, then fused WMMA GEMM
    {
        long long tot = (long long)E * 32;
        k_scatter<32, 5><<<(int)((tot + 255) / 256), 256, 0, stream>>>(x, ei, agg0, E);
        sage_gemm_relu<32, 32><<<gemm_blocks, 256, 0, stream>>>(agg0, inv, x, Wl0, Wr0, b0, h0, n);
    }
    // layer 1: aggregate h0 (64-d payload), fused WMMA GEMM with K=128
    {
        long long tot = (long long)E * 64;
        k_scatter<64, 6><<<(int)((tot + 255) / 256), 256, 0, stream>>>(h0, ei, agg1, E);
        sage_gemm_relu<64, 64><<<gemm_blocks, 256, 0, stream>>>(agg1, inv, h0, Wl1, Wr1, b1, h1, n);
    }
    // layer 2: transform-then-aggregate (64x less edge traffic)
    k_lin2   <<<(int)(((long long)n * 32 + 255) / 256), 256, 0, stream>>>(h1, Wl2, Wr2, s2l, s2r, n);
    k_scatter1<<<(E + 255) / 256, 256, 0, stream>>>(s2l, ei, s2agg, E);
    k_final  <<<(n + 255) / 256, 256, 0, stream>>>(s2agg, inv, s2r, b2, (float*)d_out, n);
}